// GINEncoder_21775484191345
// MI455X (gfx1250) — compile-verified
//
#include <hip/hip_runtime.h>
#include <hip/hip_bf16.h>

// ---------------- problem constants ----------------
#define NN    50000      // nodes
#define NE    800000     // edges
#define DH    128        // hidden / D / H / OUT
#define NL    5          // layers
#define NG    512        // graphs
#define DEMB  640        // L * H
#define BNEPS 1e-5f

typedef __attribute__((ext_vector_type(2))) float v2f;
typedef __attribute__((ext_vector_type(8))) float v8f;

// ---------------- CDNA5 async global->LDS copy ----------------
// GLOBAL_LOAD_ASYNC_TO_LDS_B128: copies 16 bytes from the global address in vaddr to the LDS address in vdst, tracked by ASYNCcnt.
// Generic LDS pointer: addr[31:0] is the wave-relative LDS offset (aperture in
// the high word), so truncation gives the VDST value.
__device__ __forceinline__ void async_copy16(void* ldsp, const void* gp) {
    asm volatile("global_load_async_to_lds_b128 %0, %1, off"
                 :: "v"((unsigned)(size_t)ldsp), "v"(gp)
                 : "memory");
}

// ---------------- small utility kernels ----------------
__global__ void zero_kernel(float* __restrict__ p, long n) {
    long i = (long)blockIdx.x * blockDim.x + threadIdx.x;
    long stride = (long)gridDim.x * blockDim.x;
    for (; i < n; i += stride) p[i] = 0.0f;
}

__global__ void copy_kernel(float* __restrict__ dst, const float* __restrict__ src, long n) {
    long i = (long)blockIdx.x * blockDim.x + threadIdx.x;
    long stride = (long)gridDim.x * blockDim.x;
    for (; i < n; i += stride) dst[i] = src[i];
}

// ---------------- edge scatter-add (segment_sum over dst) ----------------
// one wave per edge, one float4 per lane (32 lanes * 4 = 128 cols)
__global__ __launch_bounds__(256)
void scatter_kernel(const int* __restrict__ src, const int* __restrict__ dst,
                    const float* __restrict__ h, float* __restrict__ agg) {
    long gid = (long)blockIdx.x * 256 + threadIdx.x;
    if (gid >= (long)NE * 32) return;
    int e = (int)(gid >> 5);
    int l = (int)(gid & 31);
    int s = src[e];
    int d = dst[e];
    const float4 v = *(const float4*)(h + (size_t)s * DH + l * 4);
    float* p = agg + (size_t)d * DH + l * 4;
    unsafeAtomicAdd(p + 0, v.x);
    unsafeAtomicAdd(p + 1, v.y);
    unsafeAtomicAdd(p + 2, v.z);
    unsafeAtomicAdd(p + 3, v.w);
}

// h = (1 + eps[layer]) * h + agg
__global__ void combine_kernel(float* __restrict__ h, const float* __restrict__ agg,
                               const float* __restrict__ eps, int layer) {
    long i = (long)blockIdx.x * blockDim.x + threadIdx.x;
    if (i < (long)NN * DH) h[i] = (1.0f + eps[layer]) * h[i] + agg[i];
}

// ---------------- BatchNorm batch statistics ----------------
#define BN_ROWS 512
__global__ __launch_bounds__(DH)
void bn_stats_kernel(const float* __restrict__ z, float* __restrict__ sums,
                     float* __restrict__ sumsq) {
    int c = threadIdx.x;                       // column 0..127
    long r0 = (long)blockIdx.x * BN_ROWS;
    long r1 = r0 + BN_ROWS; if (r1 > NN) r1 = NN;
    float s = 0.0f, q = 0.0f;
    for (long r = r0; r < r1; ++r) {
        float v = z[r * DH + c];
        s += v;
        q += v * v;
    }
    unsafeAtomicAdd(&sums[c], s);
    unsafeAtomicAdd(&sumsq[c], q);
}

__global__ __launch_bounds__(DH)
void bn_finalize_kernel(const float* __restrict__ sums, const float* __restrict__ sumsq,
                        const float* __restrict__ g, const float* __restrict__ be,
                        float* __restrict__ scale, float* __restrict__ shift) {
    int c = threadIdx.x;
    float mean = sums[c] * (1.0f / NN);
    float var  = sumsq[c] * (1.0f / NN) - mean * mean;   // biased var (ddof=0)
    float sc   = g[c] * rsqrtf(var + BNEPS);
    scale[c] = sc;
    shift[c] = be[c] - mean * sc;
}

// v = relu(z*scale + shift); z <- v; xs[:, coloff + c] <- v
__global__ void bn_apply_kernel(float* __restrict__ z, const float* __restrict__ scale,
                                const float* __restrict__ shift,
                                float* __restrict__ xs, int coloff) {
    long i = (long)blockIdx.x * blockDim.x + threadIdx.x;
    if (i >= (long)NN * DH) return;
    int c  = (int)(i & (DH - 1));
    long n = i >> 7;
    float v = fmaxf(z[i] * scale[c] + shift[c], 0.0f);
    z[i] = v;
    xs[n * DEMB + coloff + c] = v;
}

// ---------------- graph pooling (segment_sum over batch) ----------------
__global__ __launch_bounds__(DEMB)
void pool_kernel(const float* __restrict__ xs, const int* __restrict__ batch,
                 float* __restrict__ pooled) {
    int n = blockIdx.x;
    int c = threadIdx.x;
    int g = batch[n];
    unsafeAtomicAdd(pooled + (size_t)g * DEMB + c, xs[(size_t)n * DEMB + c]);
}

// ---------------- fp32 WMMA GEMM, double-buffered staging ----------------
// C[M, Nc] = opA(A[M,K]) @ W[K,Nc] + bias, Nc = gridDim.y*64
// A tiles: async global->LDS (GLOBAL_LOAD_ASYNC_TO_LDS_B128, ASYNCcnt).
// W tiles: global->regs->LDS *transposed* (Wt[col][k], stride 38) so each
//          B fragment is a single aligned 8-byte LDS load (no repack movs).
// Inner loop software-pipelines fragment loads one k-step ahead so LDS latency
// overlaps WMMA issue instead of serializing on s_wait_dscnt 0.
// BN_A:  A' = relu(A * scale[k] + shift[k])   applied at A-fragment use
// RELU:  relu on (acc + bias)
// ACCUM: C += result  (else C = result)
// NOTE: no M-guard on A staging (async copies can't substitute zeros).
// Overflow rows (< 64 past M) read mapped workspace garbage; their accumulator
// rows are never stored. Callers guarantee the 48 rows past A are mapped.
#define TM 64
#define TN 64
#define TK 32
#define AS_STRIDE  (TK + 4)                    // 36: aligned 16B packets, clean banks
#define AS_FLOATS  (TM * AS_STRIDE)            // 2304
#define WT_STRIDE  (TK + 6)                    // 38: even (8B-aligned b64 reads)
#define WT_FLOATS  (TN * WT_STRIDE)            // 2432
#define BUF_FLOATS (AS_FLOATS + WT_FLOATS)     // 4736
#define GEMM_SMEM  (2 * BUF_FLOATS * 4)        // 37888 bytes

template<bool BN_A, bool RELU, bool ACCUM>
__global__ __launch_bounds__(128)
void gemm_wmma_kernel(const float* __restrict__ A, int ldA,
                      const float* __restrict__ W, int ldW,
                      const float* __restrict__ bias,
                      const float* __restrict__ scale,
                      const float* __restrict__ shift,
                      float* __restrict__ C, int ldC,
                      int M, int K) {
    extern __shared__ float smem[];

    const int tid  = threadIdx.x;
    const int wave = tid >> 5;
    const int lane = tid & 31;
    const int r16  = lane & 15;
    const int hi   = lane >> 4;          // 0: lanes 0-15, 1: lanes 16-31
    const int hi2  = hi << 1;
    const int row0 = blockIdx.x * TM;
    const int col0 = blockIdx.y * TN;

    float4 wreg[4];                      // W-chunk relay registers

    // issue A tile staging into smem[base..]: 4 async b128 copies per thread
    auto stageA = [&](int k0, int base) {
        #pragma unroll
        for (int it = 0; it < 4; ++it) {
            int idx = tid + it * 128;            // 0..511
            int r   = idx >> 3;                  // 0..63
            int kq  = (idx & 7) << 2;            // 0,4,...,28
            async_copy16(&smem[base + r * AS_STRIDE + kq],
                         A + (size_t)(row0 + r) * ldA + k0 + kq);
        }
    };
    // issue W chunk loads into wreg (no wait here)
    auto loadW = [&](int k0) {
        #pragma unroll
        for (int it = 0; it < 4; ++it) {
            int idx = tid + it * 128;            // 0..511
            int r   = idx >> 4;                  // k 0..31
            int cq  = (idx & 15) << 2;           // col 0,4,...,60
            wreg[it] = *(const float4*)(W + (size_t)(k0 + r) * ldW + col0 + cq);
        }
    };
    // write wreg transposed into Wt at smem[base + AS_FLOATS]
    auto storeW = [&](int base) {
        float* Wt = &smem[base + AS_FLOATS];
        #pragma unroll
        for (int it = 0; it < 4; ++it) {
            int idx = tid + it * 128;
            int r   = idx >> 4;
            int cq  = (idx & 15) << 2;
            Wt[(cq + 0) * WT_STRIDE + r] = wreg[it].x;
            Wt[(cq + 1) * WT_STRIDE + r] = wreg[it].y;
            Wt[(cq + 2) * WT_STRIDE + r] = wreg[it].z;
            Wt[(cq + 3) * WT_STRIDE + r] = wreg[it].w;
        }
    };

    v8f acc[4];
    {
        v8f z = {};
        #pragma unroll
        for (int t = 0; t < 4; ++t) acc[t] = z;
    }

    const int nChunks = K / TK;

    // prologue: fill buffer 0
    stageA(0, 0);
    loadW(0);
    storeW(0);                                        // waits loadcnt as needed
    asm volatile("s_wait_asynccnt 0x0" ::: "memory");
    __syncthreads();

    for (int c = 0; c < nChunks; ++c) {
        const int curBase = (c & 1) * BUF_FLOATS;
        const int nxtBase = BUF_FLOATS - curBase;
        const bool hasNext = (c + 1) < nChunks;

        if (hasNext) {                                // issue next chunk, no waits
            stageA((c + 1) * TK, nxtBase);
            loadW((c + 1) * TK);
        }

        const int k0 = c * TK;
        // per-chunk base pointers: lane-varying parts folded in, so the k-step
        // offset (kk*4 bytes <= 112) becomes a constant ds instruction offset
        const float* aB  = &smem[curBase + (wave * 16 + r16) * AS_STRIDE + hi2];
        const float* bB0 = &smem[curBase + AS_FLOATS + r16 * WT_STRIDE + hi2];
        const float* bB1 = bB0 + 16 * WT_STRIDE;
        const float* bB2 = bB0 + 32 * WT_STRIDE;
        const float* bB3 = bB0 + 48 * WT_STRIDE;

        // software-pipelined fragment loads: group kk+4 issues before wmmas(kk)
        v2f a0 = *(const v2f*)(aB);
        v2f b0 = *(const v2f*)(bB0);
        v2f b1 = *(const v2f*)(bB1);
        v2f b2 = *(const v2f*)(bB2);
        v2f b3 = *(const v2f*)(bB3);

        #pragma unroll
        for (int kk = 0; kk < TK; kk += 4) {
            v2f an = a0, c0 = b0, c1 = b1, c2 = b2, c3 = b3;
            if (kk + 4 < TK) {
                an = *(const v2f*)(aB  + kk + 4);
                c0 = *(const v2f*)(bB0 + kk + 4);
                c1 = *(const v2f*)(bB1 + kk + 4);
                c2 = *(const v2f*)(bB2 + kk + 4);
                c3 = *(const v2f*)(bB3 + kk + 4);
            }
            v2f au = a0;
            if (BN_A) {
                const int kg = k0 + kk + hi2;
                au.x = fmaxf(au.x * scale[kg + 0] + shift[kg + 0], 0.0f);
                au.y = fmaxf(au.y * scale[kg + 1] + shift[kg + 1], 0.0f);
            }
            acc[0] = __builtin_amdgcn_wmma_f32_16x16x4_f32(
                false, au, false, b0, (short)0, acc[0], false, false);
            acc[1] = __builtin_amdgcn_wmma_f32_16x16x4_f32(
                false, au, false, b1, (short)0, acc[1], false, false);
            acc[2] = __builtin_amdgcn_wmma_f32_16x16x4_f32(
                false, au, false, b2, (short)0, acc[2], false, false);
            acc[3] = __builtin_amdgcn_wmma_f32_16x16x4_f32(
                false, au, false, b3, (short)0, acc[3], false, false);
            a0 = an; b0 = c0; b1 = c1; b2 = c2; b3 = c3;
        }

        if (hasNext) {                                // drain + publish next buffer
            storeW(nxtBase);                          // compiler waits W loads
            asm volatile("s_wait_asynccnt 0x0" ::: "memory");
            __syncthreads();
        }
    }

    // epilogue: VGPR v, lanes 0-15 -> M=v, lanes 16-31 -> M=v+8; N = lane&15
    #pragma unroll
    for (int t = 0; t < 4; ++t) {
        #pragma unroll
        for (int vv = 0; vv < 8; ++vv) {
            int m   = vv + (hi << 3);
            int row = row0 + wave * 16 + m;
            if (row < M) {
                int col   = (t << 4) + r16;
                float val = acc[t][vv] + bias[col0 + col];
                if (RELU) val = fmaxf(val, 0.0f);
                float* cp = C + (size_t)row * ldC + col0 + col;
                if (ACCUM) *cp = *cp + val;
                else       *cp = val;
            }
        }
    }
}

// ---------------- host-side helpers ----------------
template<bool BN_A, bool RELU, bool ACCUM>
static void gemm(const float* A, int ldA, const float* W, int ldW, const float* bias,
                 const float* scale, const float* shift,
                 float* C, int ldC, int M, int K, int Nc, hipStream_t s) {
    dim3 grid((M + TM - 1) / TM, Nc / TN);
    gemm_wmma_kernel<BN_A, RELU, ACCUM><<<grid, 128, GEMM_SMEM, s>>>(
        A, ldA, W, ldW, bias, scale, shift, C, ldC, M, K);
}

static void zero(float* p, long n, hipStream_t s) {
    long blocks = (n + 255) / 256;
    if (blocks > 4096) blocks = 4096;
    zero_kernel<<<(int)blocks, 256, 0, s>>>(p, n);
}

extern "C" void kernel_launch(void* const* d_in, const int* in_sizes, int n_in,
                              void* d_out, int out_size, void* d_ws, size_t ws_size,
                              hipStream_t stream) {
    // ---- inputs (setup_inputs order) ----
    const float* x      = (const float*)d_in[0];
    const int*   edge   = (const int*)d_in[1];   // [2, E]
    const int*   batch  = (const int*)d_in[2];   // [N]
    const float* cW1    = (const float*)d_in[3];
    const float* cb1    = (const float*)d_in[4];
    const float* cg1    = (const float*)d_in[5];
    const float* cbe1   = (const float*)d_in[6];
    const float* cW2    = (const float*)d_in[7];
    const float* cb2    = (const float*)d_in[8];
    const float* eps    = (const float*)d_in[9];
    const float* bng    = (const float*)d_in[10];
    const float* bnb    = (const float*)d_in[11];
    const float* predW  = (const float*)d_in[12];
    const float* predb  = (const float*)d_in[13];
    const float* gW     = (const float*)d_in[14];
    const float* gb     = (const float*)d_in[15];
    const float* gsW    = (const float*)d_in[16];
    const float* gsb    = (const float*)d_in[17];
    const float* lW     = (const float*)d_in[18];
    const float* lb     = (const float*)d_in[19];
    const float* lsW    = (const float*)d_in[20];
    const float* lsb    = (const float*)d_in[21];

    const int* esrc = edge;
    const int* edst = edge + NE;

    // ---- workspace layout (floats) ----
    // NOTE: ordering matters: every GEMM A-operand region must have >= 48*640
    // mapped floats after it (async staging reads up to 48 rows past M).
    float* ws = (float*)d_ws;
    size_t o = 0;
    float* h      = ws + o; o += (size_t)NN * DH;     // node features
    float* z1     = ws + o; o += (size_t)NN * DH;     // agg, then z1
    float* xs     = ws + o; o += (size_t)NN * DEMB;   // concat(xs) [N, 640]
    float* nt1    = ws + o; o += (size_t)NN * DEMB;   // node-FF ping
    float* nt2    = ws + o; o += (size_t)NN * DEMB;   // node-FF pong
    float* pooled = ws + o; o += (size_t)NG * DEMB;
    float* gt1    = ws + o; o += (size_t)NG * DEMB;
    float* gt2    = ws + o; o += (size_t)NG * DEMB;
    float* stats  = ws + o; o += 512;                 // sums|sumsq|scale|shift
    float* ssum   = stats;
    float* ssq    = stats + 128;
    float* sscale = stats + 256;
    float* sshift = stats + 384;

    // ---- outputs: (graph_embedding [G,640], node_embedding [N,640], xcat [G,640]) ----
    float* gout = (float*)d_out;
    float* nout = gout + (size_t)NG * DEMB;
    float* xcat = nout + (size_t)NN * DEMB;

    // h = x
    copy_kernel<<<4096, 256, 0, stream>>>(h, x, (long)NN * DH);

    const long elem = (long)NN * DH;
    const int elemBlocks = (int)((elem + 255) / 256);
    const int scatBlocks = (int)(((long)NE * 32 + 255) / 256);
    const int bnBlocks   = (NN + BN_ROWS - 1) / BN_ROWS;

    for (int i = 0; i < NL; ++i) {
        // agg = segment_sum(h[src] -> dst)
        zero(z1, elem, stream);
        scatter_kernel<<<scatBlocks, 256, 0, stream>>>(esrc, edst, h, z1);
        // h = (1+eps)*h + agg
        combine_kernel<<<elemBlocks, 256, 0, stream>>>(h, z1, eps, i);
        // z1 = h @ W1_i + b1_i
        gemm<false, false, false>(h, DH, cW1 + (size_t)i * DH * DH, DH, cb1 + i * DH,
                                  nullptr, nullptr, z1, DH, NN, DH, DH, stream);
        // inner BN stats on z1 -> scale/shift
        zero(stats, 256, stream);
        bn_stats_kernel<<<bnBlocks, DH, 0, stream>>>(z1, ssum, ssq);
        bn_finalize_kernel<<<1, DH, 0, stream>>>(ssum, ssq, cg1 + i * DH, cbe1 + i * DH,
                                                 sscale, sshift);
        // h = relu(BN(z1)) @ W2_i + b2_i      (BN+ReLU fused into A-fragment use)
        gemm<true, false, false>(z1, DH, cW2 + (size_t)i * DH * DH, DH, cb2 + i * DH,
                                 sscale, sshift, h, DH, NN, DH, DH, stream);
        // outer BN stats on h
        zero(stats, 256, stream);
        bn_stats_kernel<<<bnBlocks, DH, 0, stream>>>(h, ssum, ssq);
        bn_finalize_kernel<<<1, DH, 0, stream>>>(ssum, ssq, bng + i * DH, bnb + i * DH,
                                                 sscale, sshift);
        // h = relu(BN(h)); xs[:, i*128:(i+1)*128] = h
        bn_apply_kernel<<<elemBlocks, 256, 0, stream>>>(h, sscale, sshift, xs, i * DH);
    }

    // pooled[g] = sum over nodes of xs
    zero(pooled, (long)NG * DEMB, stream);
    pool_kernel<<<NN, DEMB, 0, stream>>>(xs, batch, pooled);

    // prediction heads -> xcat (written directly into d_out slice, reused as FF input)
    for (int i = 0; i < NL; ++i) {
        gemm<false, false, false>(pooled + i * DH, DEMB,
                                  predW + (size_t)i * DH * DH, DH, predb + i * DH,
                                  nullptr, nullptr, xcat + i * DH, DEMB,
                                  NG, DH, DH, stream);
    }

    // graph FF: gout = relu3-chain(xcat) + xcat @ gsW + gsb
    gemm<false, false, false>(xcat, DEMB, gsW, DEMB, gsb, nullptr, nullptr,
                              gout, DEMB, NG, DEMB, DEMB, stream);
    gemm<false, true, false>(xcat, DEMB, gW, DEMB, gb, nullptr, nullptr,
                             gt1, DEMB, NG, DEMB, DEMB, stream);
    gemm<false, true, false>(gt1, DEMB, gW + (size_t)DEMB * DEMB, DEMB, gb + DEMB,
                             nullptr, nullptr, gt2, DEMB, NG, DEMB, DEMB, stream);
    gemm<false, true, true>(gt2, DEMB, gW + 2 * (size_t)DEMB * DEMB, DEMB, gb + 2 * DEMB,
                            nullptr, nullptr, gout, DEMB, NG, DEMB, DEMB, stream);

    // node FF: nout = relu3-chain(xs) + xs @ lsW + lsb
    gemm<false, false, false>(xs, DEMB, lsW, DEMB, lsb, nullptr, nullptr,
                              nout, DEMB, NN, DEMB, DEMB, stream);
    gemm<false, true, false>(xs, DEMB, lW, DEMB, lb, nullptr, nullptr,
                             nt1, DEMB, NN, DEMB, DEMB, stream);
    gemm<false, true, false>(nt1, DEMB, lW + (size_t)DEMB * DEMB, DEMB, lb + DEMB,
                             nullptr, nullptr, nt2, DEMB, NN, DEMB, DEMB, stream);
    gemm<false, true, true>(nt2, DEMB, lW + 2 * (size_t)DEMB * DEMB, DEMB, lb + 2 * DEMB,
                            nullptr, nullptr, nout, DEMB, NN, DEMB, DEMB, stream);
}